// BiAttention_41188736368724
// MI455X (gfx1250) — compile-verified
//
#include <hip/hip_runtime.h>
#include <math.h>

// Problem constants
constexpr int NB = 32;    // batch
constexpr int L  = 1024;  // Lp == Lh
constexpr int DD = 1024;  // feature dim
constexpr int KC = 16;    // K-chunk staged in LDS

// LDS row pitches (dwords) chosen for bank-conflict-free WMMA fragment reads
constexpr int PITCH_MK = 19;   // [row][k] tiles: 16 dwords + 3 pad (TDM pad codes 3/2)
constexpr int PITCH_KN = 257;  // [k][n] tiles: 256 dwords + 1 pad (TDM pad codes 7/0)
constexpr int PITCH_KM = 73;   // [k][m] transposed softmax tiles (manual staging)

typedef float v2f __attribute__((ext_vector_type(2)));
typedef float v8f __attribute__((ext_vector_type(8)));
typedef unsigned int v4u __attribute__((ext_vector_type(4)));
typedef int v4i __attribute__((ext_vector_type(4)));
typedef int v8i __attribute__((ext_vector_type(8)));

// Low 32 bits of a flat shared-memory address are the LDS byte offset.
__device__ __forceinline__ unsigned lds_off(const void* p) {
    return (unsigned)(size_t)p;
}

// Issue a TDM 2D tile load: tileRows rows of tileW fp32 elements, row stride
// strideElems, destination LDS with padAmount dwords inserted every
// padInterval dwords (codes per D# group1 encoding).
__device__ __forceinline__ void tdm_load_2d(unsigned ldsAddr, const float* gptr,
                                            unsigned tileW, unsigned tileRows,
                                            unsigned strideElems,
                                            unsigned padIntervalCode,
                                            unsigned padAmountCode) {
    unsigned long long ga = (unsigned long long)(size_t)gptr;
    v4u g0;
    g0.x = 1u;                                     // count=1 (valid descriptor)
    g0.y = ldsAddr;                                // lds_addr (bytes)
    g0.z = (unsigned)ga;                           // global_addr[31:0]
    g0.w = (unsigned)((ga >> 32) & 0x01FFFFFFu) |  // global_addr[56:32]
           (2u << 30);                             // type = 2 ("image")
    v8i g1;
    g1[0] = (int)((2u << 16) |                     // data_size = 2 (4 bytes)
                  (1u << 20) |                     // pad_enable
                  (padIntervalCode << 22) |
                  (padAmountCode << 25));
    g1[1] = (int)((tileW & 0xFFFFu) << 16);                          // tensor_dim0 lo16
    g1[2] = (int)((tileW >> 16) | ((tileRows & 0xFFFFu) << 16));     // dim0 hi | dim1 lo
    g1[3] = (int)((tileRows >> 16) | ((tileW & 0xFFFFu) << 16));     // dim1 hi | tile_dim0
    g1[4] = (int)(tileRows & 0xFFFFu);                               // tile_dim1 (tile_dim2=0)
    g1[5] = (int)strideElems;                                        // tensor_dim0_stride lo32
    g1[6] = 0;
    g1[7] = 0;
    v4i g2 = {0, 0, 0, 0};
    v4i g3 = {0, 0, 0, 0};
    v8i g4 = {0, 0, 0, 0, 0, 0, 0, 0};
    __builtin_amdgcn_tensor_load_to_lds(g0, g1, g2, g3, g4, 0);
}

// ---------------------------------------------------------------------------
// Kernel 1: S[b,p,h] = sum_k P[b,p,k] * H[b,h,k]
// WG = 256 threads (8 waves, 2Mx4N), WG tile 64x256, wave tile 32x64.
// Both tiles DMA'd by TDM (wave 0), double-buffered LDS, 1 barrier/chunk.
// ---------------------------------------------------------------------------
__global__ __launch_bounds__(256) void k_scores(const float* __restrict__ P,
                                                const float* __restrict__ H,
                                                float* __restrict__ S) {
    __shared__ float As[2][64][PITCH_MK];    // A[m][k]
    __shared__ float Bs[2][256][PITCH_MK];   // B[n][k] = H[n][k]

    const int b     = blockIdx.z;
    const int pTile = blockIdx.y * 64;
    const int hTile = blockIdx.x * 256;
    const int tid   = threadIdx.x;
    const int lane  = tid & 31;
    const int wave  = tid >> 5;
    const int waveM = wave >> 2;          // 0..1
    const int waveN = wave & 3;           // 0..3
    const int mlane = lane & 15;
    const int khalf = (lane >> 4) * 2;

    const float* Pb = P + (size_t)b * L * DD;
    const float* Hb = H + (size_t)b * L * DD;

    v8f acc[2][4];
#pragma unroll
    for (int i = 0; i < 2; ++i)
#pragma unroll
        for (int j = 0; j < 4; ++j)
#pragma unroll
            for (int e = 0; e < 8; ++e) acc[i][j][e] = 0.0f;

    if (wave == 0) {
        tdm_load_2d(lds_off(&As[0][0][0]), Pb + (size_t)pTile * DD, KC, 64, DD, 3, 2);
        tdm_load_2d(lds_off(&Bs[0][0][0]), Hb + (size_t)hTile * DD, KC, 256, DD, 3, 2);
        __builtin_amdgcn_s_wait_tensorcnt(0);
    }
    __syncthreads();

    const int NCH = DD / KC;   // 64 chunks
    for (int c = 0; c < NCH; ++c) {
        const int cur = c & 1;
        const int nxt = cur ^ 1;
        if (wave == 0 && c + 1 < NCH) {
            const int kB = (c + 1) * KC;
            tdm_load_2d(lds_off(&As[nxt][0][0]), Pb + (size_t)pTile * DD + kB, KC, 64, DD, 3, 2);
            tdm_load_2d(lds_off(&Bs[nxt][0][0]), Hb + (size_t)hTile * DD + kB, KC, 256, DD, 3, 2);
        }
#pragma unroll
        for (int kb = 0; kb < KC; kb += 4) {
            v2f a[2], bb[4];
#pragma unroll
            for (int i = 0; i < 2; ++i) {
                const int m = waveM * 32 + i * 16 + mlane;
                a[i].x = As[cur][m][kb + khalf];
                a[i].y = As[cur][m][kb + khalf + 1];
            }
#pragma unroll
            for (int j = 0; j < 4; ++j) {
                const int n = waveN * 64 + j * 16 + mlane;
                bb[j].x = Bs[cur][n][kb + khalf];
                bb[j].y = Bs[cur][n][kb + khalf + 1];
            }
#pragma unroll
            for (int i = 0; i < 2; ++i)
#pragma unroll
                for (int j = 0; j < 4; ++j)
                    acc[i][j] = __builtin_amdgcn_wmma_f32_16x16x4_f32(
                        false, a[i], false, bb[j], (short)0, acc[i][j], false, false);
        }
        if (wave == 0) __builtin_amdgcn_s_wait_tensorcnt(0);
        __syncthreads();
    }

    float* Sb = S + (size_t)b * L * L;
#pragma unroll
    for (int i = 0; i < 2; ++i) {
        int prow = pTile + waveM * 32 + i * 16 + (lane >> 4) * 8;
#pragma unroll
        for (int j = 0; j < 4; ++j) {
            int hcol = hTile + waveN * 64 + j * 16 + mlane;
#pragma unroll
            for (int r = 0; r < 8; ++r)
                Sb[(size_t)(prow + r) * L + hcol] = acc[i][j][r];
        }
    }
}

// ---------------------------------------------------------------------------
// Kernel 2: per-row (over h, masked h < hyp_len[b]) online max & inv-sum.
// ---------------------------------------------------------------------------
__global__ __launch_bounds__(256) void k_rowstats(const float* __restrict__ S,
                                                  const int* __restrict__ hyp_lens,
                                                  float* __restrict__ rowmax,
                                                  float* __restrict__ rowinv) {
    const int rp   = blockIdx.x * 8 + (threadIdx.x >> 5);
    const int lane = threadIdx.x & 31;
    const int b    = rp >> 10;
    const int hlen = hyp_lens[b];
    const float* row = S + (size_t)rp * L;

    float m = -INFINITY, s = 0.0f;
    for (int h = lane; h < hlen; h += 32) {
        float v  = row[h];
        float nm = fmaxf(m, v);
        s = s * expf(m - nm) + expf(v - nm);
        m = nm;
    }
#pragma unroll
    for (int off = 16; off > 0; off >>= 1) {
        float om = __shfl_xor(m, off, 32);
        float os = __shfl_xor(s, off, 32);
        float M  = fmaxf(m, om);
        float w0 = (m  == -INFINITY) ? 0.0f : expf(m - M);
        float w1 = (om == -INFINITY) ? 0.0f : expf(om - M);
        s = s * w0 + os * w1;
        m = M;
    }
    if (lane == 0) {
        rowmax[rp] = m;
        rowinv[rp] = 1.0f / s;
    }
}

// ---------------------------------------------------------------------------
// Kernel 3: per-column (over p, masked p < prem_len[b]) online max & inv-sum.
// ---------------------------------------------------------------------------
__global__ __launch_bounds__(256) void k_colstats(const float* __restrict__ S,
                                                  const int* __restrict__ prem_lens,
                                                  float* __restrict__ colmax,
                                                  float* __restrict__ colinv) {
    const int b    = blockIdx.x >> 2;
    const int h    = ((blockIdx.x & 3) * 256) + threadIdx.x;
    const int plen = prem_lens[b];
    const float* Sb = S + (size_t)b * L * L;

    float m = -INFINITY, s = 0.0f;
    for (int p = 0; p < plen; ++p) {
        float v  = Sb[(size_t)p * L + h];
        float nm = fmaxf(m, v);
        s = s * expf(m - nm) + expf(v - nm);
        m = nm;
    }
    colmax[b * L + h] = m;
    colinv[b * L + h] = 1.0f / s;
}

// ---------------------------------------------------------------------------
// Kernel 4: ctx1 = rowsoftmax(S) @ H; epilogue r1[p,b,d]=P-ctx1,
// r1[p,b,D+d]=ctx1*P (layout [Lp, B, 2D]).
// A tile register-staged (softmax applied), B tile via TDM; double-buffered.
// ---------------------------------------------------------------------------
__global__ __launch_bounds__(256) void k_ctx1(const float* __restrict__ S,
                                              const float* __restrict__ H,
                                              const float* __restrict__ P,
                                              const float* __restrict__ rowmax,
                                              const float* __restrict__ rowinv,
                                              const int* __restrict__ hyp_lens,
                                              float* __restrict__ out1) {
    __shared__ float As[2][KC][PITCH_KM];    // A[k=h][m=p]
    __shared__ float Bs[2][KC][PITCH_KN];    // B[k=h][n=d]

    const int b     = blockIdx.z;
    const int pTile = blockIdx.y * 64;
    const int dTile = blockIdx.x * 256;
    const int tid   = threadIdx.x;
    const int lane  = tid & 31;
    const int wave  = tid >> 5;
    const int waveM = wave >> 2;
    const int waveN = wave & 3;
    const int mlane = lane & 15;
    const int khalf = (lane >> 4) * 2;
    const int hlen  = hyp_lens[b];

    const float* Sb = S + (size_t)b * L * L;
    const float* Hb = H + (size_t)b * L * DD;
    const float* Pb = P + (size_t)b * L * DD;

    v8f acc[2][4];
#pragma unroll
    for (int i = 0; i < 2; ++i)
#pragma unroll
        for (int j = 0; j < 4; ++j)
#pragma unroll
            for (int e = 0; e < 8; ++e) acc[i][j][e] = 0.0f;

    float regA[4];
    auto loadA = [&](int kBase) {
#pragma unroll
        for (int r = 0; r < 4; ++r) {
            int idx = tid + r * 256;
            int k = idx & 15, m = idx >> 4;
            int hh = kBase + k;
            int pg = pTile + m;
            float v = Sb[(size_t)pg * L + hh];
            regA[r] = (hh < hlen) ? expf(v - rowmax[b * L + pg]) * rowinv[b * L + pg] : 0.0f;
        }
    };
    auto storeA = [&](int buf) {
#pragma unroll
        for (int r = 0; r < 4; ++r) {
            int idx = tid + r * 256;
            int k = idx & 15, m = idx >> 4;
            As[buf][k][m] = regA[r];
        }
    };

    loadA(0);
    if (wave == 0)
        tdm_load_2d(lds_off(&Bs[0][0][0]), Hb + dTile, 256, KC, DD, 7, 0);
    storeA(0);
    if (wave == 0) __builtin_amdgcn_s_wait_tensorcnt(0);
    __syncthreads();

    const int NCH = L / KC;   // 64 chunks (K = h)
    for (int c = 0; c < NCH; ++c) {
        const int cur = c & 1;
        const int nxt = cur ^ 1;
        if (c + 1 < NCH) {
            loadA((c + 1) * KC);
            if (wave == 0)
                tdm_load_2d(lds_off(&Bs[nxt][0][0]),
                            Hb + (size_t)(c + 1) * KC * DD + dTile, 256, KC, DD, 7, 0);
        }
#pragma unroll
        for (int kb = 0; kb < KC; kb += 4) {
            v2f a[2], bb[4];
#pragma unroll
            for (int i = 0; i < 2; ++i) {
                const int m = waveM * 32 + i * 16 + mlane;
                a[i].x = As[cur][kb + khalf][m];
                a[i].y = As[cur][kb + khalf + 1][m];
            }
#pragma unroll
            for (int j = 0; j < 4; ++j) {
                const int n = waveN * 64 + j * 16 + mlane;
                bb[j].x = Bs[cur][kb + khalf][n];
                bb[j].y = Bs[cur][kb + khalf + 1][n];
            }
#pragma unroll
            for (int i = 0; i < 2; ++i)
#pragma unroll
                for (int j = 0; j < 4; ++j)
                    acc[i][j] = __builtin_amdgcn_wmma_f32_16x16x4_f32(
                        false, a[i], false, bb[j], (short)0, acc[i][j], false, false);
        }
        if (c + 1 < NCH) storeA(nxt);
        if (wave == 0) __builtin_amdgcn_s_wait_tensorcnt(0);
        __syncthreads();
    }

#pragma unroll
    for (int i = 0; i < 2; ++i) {
        int prow = pTile + waveM * 32 + i * 16 + (lane >> 4) * 8;
#pragma unroll
        for (int j = 0; j < 4; ++j) {
            int dcol = dTile + waveN * 64 + j * 16 + mlane;
#pragma unroll
            for (int r = 0; r < 8; ++r) {
                int p = prow + r;
                float c  = acc[i][j][r];
                float pv = Pb[(size_t)p * DD + dcol];
                size_t o = ((size_t)p * NB + b) * (2 * DD) + dcol;
                out1[o]      = pv - c;
                out1[o + DD] = c * pv;
            }
        }
    }
}

// ---------------------------------------------------------------------------
// Kernel 5: ctx2 = colsoftmax(S)^T @ P; epilogue r2[b,h,d]=H-ctx2,
// r2[b,h,D+d]=ctx2*H (layout [B, Lh, 2D]).
// ---------------------------------------------------------------------------
__global__ __launch_bounds__(256) void k_ctx2(const float* __restrict__ S,
                                              const float* __restrict__ P,
                                              const float* __restrict__ H,
                                              const float* __restrict__ colmax,
                                              const float* __restrict__ colinv,
                                              const int* __restrict__ prem_lens,
                                              float* __restrict__ out2) {
    __shared__ float As[2][KC][PITCH_KM];    // A[k=p][m=h]
    __shared__ float Bs[2][KC][PITCH_KN];    // B[k=p][n=d]

    const int b     = blockIdx.z;
    const int hTile = blockIdx.y * 64;
    const int dTile = blockIdx.x * 256;
    const int tid   = threadIdx.x;
    const int lane  = tid & 31;
    const int wave  = tid >> 5;
    const int waveM = wave >> 2;
    const int waveN = wave & 3;
    const int mlane = lane & 15;
    const int khalf = (lane >> 4) * 2;
    const int plen  = prem_lens[b];

    const float* Sb = S + (size_t)b * L * L;
    const float* Pb = P + (size_t)b * L * DD;
    const float* Hb = H + (size_t)b * L * DD;

    v8f acc[2][4];
#pragma unroll
    for (int i = 0; i < 2; ++i)
#pragma unroll
        for (int j = 0; j < 4; ++j)
#pragma unroll
            for (int e = 0; e < 8; ++e) acc[i][j][e] = 0.0f;

    float regA[4];
    auto loadA = [&](int kBase) {
#pragma unroll
        for (int r = 0; r < 4; ++r) {
            int idx = tid + r * 256;
            int m = idx & 63, k = idx >> 6;
            int pg = kBase + k;
            int hg = hTile + m;
            float v = Sb[(size_t)pg * L + hg];
            regA[r] = (pg < plen) ? expf(v - colmax[b * L + hg]) * colinv[b * L + hg] : 0.0f;
        }
    };
    auto storeA = [&](int buf) {
#pragma unroll
        for (int r = 0; r < 4; ++r) {
            int idx = tid + r * 256;
            int m = idx & 63, k = idx >> 6;
            As[buf][k][m] = regA[r];
        }
    };

    loadA(0);
    if (wave == 0)
        tdm_load_2d(lds_off(&Bs[0][0][0]), Pb + dTile, 256, KC, DD, 7, 0);
    storeA(0);
    if (wave == 0) __builtin_amdgcn_s_wait_tensorcnt(0);
    __syncthreads();

    const int NCH = L / KC;   // 64 chunks (K = p)
    for (int c = 0; c < NCH; ++c) {
        const int cur = c & 1;
        const int nxt = cur ^ 1;
        if (c + 1 < NCH) {
            loadA((c + 1) * KC);
            if (wave == 0)
                tdm_load_2d(lds_off(&Bs[nxt][0][0]),
                            Pb + (size_t)(c + 1) * KC * DD + dTile, 256, KC, DD, 7, 0);
        }
#pragma unroll
        for (int kb = 0; kb < KC; kb += 4) {
            v2f a[2], bb[4];
#pragma unroll
            for (int i = 0; i < 2; ++i) {
                const int m = waveM * 32 + i * 16 + mlane;
                a[i].x = As[cur][kb + khalf][m];
                a[i].y = As[cur][kb + khalf + 1][m];
            }
#pragma unroll
            for (int j = 0; j < 4; ++j) {
                const int n = waveN * 64 + j * 16 + mlane;
                bb[j].x = Bs[cur][kb + khalf][n];
                bb[j].y = Bs[cur][kb + khalf + 1][n];
            }
#pragma unroll
            for (int i = 0; i < 2; ++i)
#pragma unroll
                for (int j = 0; j < 4; ++j)
                    acc[i][j] = __builtin_amdgcn_wmma_f32_16x16x4_f32(
                        false, a[i], false, bb[j], (short)0, acc[i][j], false, false);
        }
        if (c + 1 < NCH) storeA(nxt);
        if (wave == 0) __builtin_amdgcn_s_wait_tensorcnt(0);
        __syncthreads();
    }

#pragma unroll
    for (int i = 0; i < 2; ++i) {
        int hrow = hTile + waveM * 32 + i * 16 + (lane >> 4) * 8;
#pragma unroll
        for (int j = 0; j < 4; ++j) {
            int dcol = dTile + waveN * 64 + j * 16 + mlane;
#pragma unroll
            for (int r = 0; r < 8; ++r) {
                int h = hrow + r;
                float c  = acc[i][j][r];
                float hv = Hb[(size_t)h * DD + dcol];
                size_t o = ((size_t)b * L + h) * (2 * DD) + dcol;
                out2[o]      = hv - c;
                out2[o + DD] = c * hv;
            }
        }
    }
}

// ---------------------------------------------------------------------------
extern "C" void kernel_launch(void* const* d_in, const int* in_sizes, int n_in,
                              void* d_out, int out_size, void* d_ws, size_t ws_size,
                              hipStream_t stream) {
    const float* premise    = (const float*)d_in[0];
    const int*   prem_lens  = (const int*)d_in[1];
    const float* hypothesis = (const float*)d_in[2];
    const int*   hyp_lens   = (const int*)d_in[3];
    float* out = (float*)d_out;

    float* S      = (float*)d_ws;                  // 128 MB, L2-resident
    float* rowmax = S + (size_t)NB * L * L;
    float* rowinv = rowmax + NB * L;
    float* colmax = rowinv + NB * L;
    float* colinv = colmax + NB * L;

    float* out1 = out;                             // [Lp, B, 2D]
    float* out2 = out + (size_t)L * NB * 2 * DD;   // [B, Lh, 2D]

    dim3 blk(256);
    k_scores  <<<dim3(L / 256, L / 64, NB), blk, 0, stream>>>(premise, hypothesis, S);
    k_rowstats<<<dim3(NB * L / 8),          blk, 0, stream>>>(S, hyp_lens, rowmax, rowinv);
    k_colstats<<<dim3(NB * 4),              blk, 0, stream>>>(S, prem_lens, colmax, colinv);
    k_ctx1    <<<dim3(DD / 256, L / 64, NB), blk, 0, stream>>>(S, hypothesis, premise,
                                                               rowmax, rowinv, hyp_lens, out1);
    k_ctx2    <<<dim3(DD / 256, L / 64, NB), blk, 0, stream>>>(S, premise, hypothesis,
                                                               colmax, colinv, prem_lens, out2);
}